// CoLa_85950885527989
// MI455X (gfx1250) — compile-verified
//
#include <hip/hip_runtime.h>
#include <math.h>

// Problem constants (must match reference)
#define B_SZ   256
#define IN_F   1024
#define OUT_F  1024
#define SUB_F  128
#define Q_F    1024
#define N_SUB  64
#define N_ACT  8
#define N_ASSIGN (B_SZ * N_ACT)   // 2048 (b, subnet) assignments

typedef float v2f __attribute__((ext_vector_type(2)));
typedef float v8f __attribute__((ext_vector_type(8)));

// ---------------------------------------------------------------------------
// K0: zero output + bucket counters
// ---------------------------------------------------------------------------
__global__ void k_zero(float* __restrict__ y, int ny,
                       int* __restrict__ cnt_cur, int nc) {
    int i = blockIdx.x * blockDim.x + threadIdx.x;
    if (i < ny) y[i] = 0.0f;
    if (i < nc) cnt_cur[i] = 0;
}

// ---------------------------------------------------------------------------
// K1: attention = q @ Wk^T + bk ; top-8 by rank ; softmax over the 8.
// One block per batch row, one thread per subnet (64).
// ---------------------------------------------------------------------------
__global__ __launch_bounds__(N_SUB)
void k_attn_topk(const float* __restrict__ q, const float* __restrict__ Wk,
                 const float* __restrict__ bk,
                 int* __restrict__ act_s, float* __restrict__ act_w,
                 int* __restrict__ counts) {
    const int b = blockIdx.x;
    const int s = threadIdx.x;

    const float4* q4 = reinterpret_cast<const float4*>(q + (size_t)b * Q_F);
    const float4* w4 = reinterpret_cast<const float4*>(Wk + (size_t)s * Q_F);
    float acc = bk[s];
#pragma unroll 4
    for (int i = 0; i < Q_F / 4; ++i) {
        float4 a = q4[i], c = w4[i];
        acc += a.x * c.x + a.y * c.y + a.z * c.z + a.w * c.w;
    }

    __shared__ float satt[N_SUB];
    __shared__ float sexp[N_SUB];
    satt[s] = acc;
    __syncthreads();

    // Rank = number of entries strictly greater (ties broken by index).
    // Exactly N_ACT entries have rank < N_ACT.
    int rank = 0;
    float mx = -3.402823466e38f;
    for (int t = 0; t < N_SUB; ++t) {
        float v = satt[t];
        mx = fmaxf(mx, v);
        rank += (v > acc) || (v == acc && t < s);
    }
    float e = (rank < N_ACT) ? expf(acc - mx) : 0.0f;
    sexp[s] = e;
    __syncthreads();

    float ssum = 0.0f;
    for (int t = 0; t < N_SUB; ++t) ssum += sexp[t];   // deterministic order

    if (rank < N_ACT) {
        act_s[b * N_ACT + rank] = s;
        act_w[b * N_ACT + rank] = e / ssum;
        atomicAdd(&counts[s], 1);
    }
}

// ---------------------------------------------------------------------------
// K2: exclusive scan of bucket counts + scatter assignments into buckets.
// Single block (work is tiny: 64 counts, 2048 assignments).
// ---------------------------------------------------------------------------
__global__ __launch_bounds__(256)
void k_scan_scatter(const int* __restrict__ counts, int* __restrict__ offsets,
                    int* __restrict__ cursor,
                    const int* __restrict__ act_s, const float* __restrict__ act_w,
                    int* __restrict__ row_b, float* __restrict__ row_w) {
    __shared__ int soff[N_SUB];
    const int tid = threadIdx.x;
    if (tid == 0) {
        int acc = 0;
        for (int s = 0; s < N_SUB; ++s) { soff[s] = acc; acc += counts[s]; }
    }
    __syncthreads();
    if (tid < N_SUB) offsets[tid] = soff[tid];

    for (int t = tid; t < N_ASSIGN; t += blockDim.x) {
        int s = act_s[t];
        int pos = soff[s] + atomicAdd(&cursor[s], 1);
        row_b[pos] = t >> 3;            // batch row
        row_w[pos] = act_w[t];          // softmax weight
    }
}

// ---------------------------------------------------------------------------
// K3 (stage A): per subnet s, H_s = X_gather @ V0[s]^T, weight-scaled rows.
//   H[pos, r] = w[pos] * sum_i x[row_b[pos], i] * V0[s, r, i]
// grid = (SUB_F/16 n-tiles, 16 m-tiles, N_SUB), one wave32 per 16x16 tile.
// ---------------------------------------------------------------------------
__global__ __launch_bounds__(32)
void k_stageA(const float* __restrict__ x, const float* __restrict__ V0,
              const int* __restrict__ counts, const int* __restrict__ offsets,
              const int* __restrict__ row_b, const float* __restrict__ row_w,
              float* __restrict__ H) {
    const int s  = blockIdx.z;
    const int mt = blockIdx.y;
    const int nt = blockIdx.x;
    const int cnt = counts[s];
    if (mt * 16 >= cnt) return;                       // uniform per block
    const int off  = offsets[s];
    const int rem  = cnt - mt * 16;                   // valid rows in tile

    const int lane = threadIdx.x;
    const int half = lane >> 4;                       // K sub-group (0/1)
    const int lm   = lane & 15;                       // M (for A) / N (for B)

    // A fragment source: gathered batch row (clamped for padded rows)
    const int rowA = row_b[off + mt * 16 + ((lm < rem) ? lm : 0)];
    const float* ar = x  + (size_t)rowA * IN_F + 2 * half;
    // B fragment source: V0[s] row r = nt*16 + lm  (B[k][n] = V0[s][n][k])
    const float* br = V0 + (size_t)s * SUB_F * IN_F
                         + (size_t)(nt * 16 + lm) * IN_F + 2 * half;

    v8f c = {0.f, 0.f, 0.f, 0.f, 0.f, 0.f, 0.f, 0.f};
#pragma unroll 8
    for (int k = 0; k < IN_F; k += 4) {
        v2f a = *reinterpret_cast<const v2f*>(ar + k);
        v2f b = *reinterpret_cast<const v2f*>(br + k);
        c = __builtin_amdgcn_wmma_f32_16x16x4_f32(false, a, false, b,
                                                  (short)0, c, false, false);
    }

    // C/D layout: VGPR v -> M = v + 8*half, N = lm
    float* hp = H + (size_t)(off + mt * 16) * SUB_F + nt * 16 + lm;
    if (rem >= 16) {
        // full tile: unconditional stores
#pragma unroll
        for (int v = 0; v < 8; ++v) {
            int ml = v + 8 * half;
            hp[(size_t)ml * SUB_F] = c[v] * row_w[off + mt * 16 + ml];
        }
    } else {
#pragma unroll
        for (int v = 0; v < 8; ++v) {
            int ml = v + 8 * half;
            if (ml < rem)
                hp[(size_t)ml * SUB_F] = c[v] * row_w[off + mt * 16 + ml];
        }
    }
}

// ---------------------------------------------------------------------------
// K4 (stage B): per subnet s, Y_s = H_s @ V1[s]^T, atomic-accumulated into y.
//   y[row_b[pos], o] += sum_r H[pos, r] * V1[s, o, r]
// 32(m) x 16(o) tile per wave: the B fragment (V1, the dominant data) is
// loaded once and shared by two WMMAs -> 3 loads / 2 WMMA.
// grid = (OUT_F/16 o-tiles, B/32 m-tiles, N_SUB).
// ---------------------------------------------------------------------------
__global__ __launch_bounds__(32)
void k_stageB(const float* __restrict__ H, const float* __restrict__ V1,
              const int* __restrict__ counts, const int* __restrict__ offsets,
              const int* __restrict__ row_b, float* __restrict__ y) {
    const int s  = blockIdx.z;
    const int mt = blockIdx.y;                        // 32-row tiles
    const int ot = blockIdx.x;
    const int cnt = counts[s];
    if (mt * 32 >= cnt) return;
    const int off  = offsets[s];
    const int rem  = cnt - mt * 32;                   // 1..32 valid rows

    const int lane = threadIdx.x;
    const int half = lane >> 4;
    const int lm   = lane & 15;

    // Two A fragments (rows [0,16) and [16,32) of the tile), clamped pads
    const int p0 = off + mt * 32 + ((lm      < rem) ? lm      : 0);
    const int p1 = off + mt * 32 + ((lm + 16 < rem) ? lm + 16 : 0);
    const float* a0 = H  + (size_t)p0 * SUB_F + 2 * half;
    const float* a1 = H  + (size_t)p1 * SUB_F + 2 * half;
    const float* br = V1 + (size_t)s * OUT_F * SUB_F
                         + (size_t)(ot * 16 + lm) * SUB_F + 2 * half;

    v8f c0 = {0.f, 0.f, 0.f, 0.f, 0.f, 0.f, 0.f, 0.f};
    v8f c1 = {0.f, 0.f, 0.f, 0.f, 0.f, 0.f, 0.f, 0.f};
#pragma unroll 8
    for (int k = 0; k < SUB_F; k += 4) {
        v2f b  = *reinterpret_cast<const v2f*>(br + k);
        v2f a  = *reinterpret_cast<const v2f*>(a0 + k);
        v2f a2 = *reinterpret_cast<const v2f*>(a1 + k);
        c0 = __builtin_amdgcn_wmma_f32_16x16x4_f32(false, a,  false, b,
                                                   (short)0, c0, false, false);
        c1 = __builtin_amdgcn_wmma_f32_16x16x4_f32(false, a2, false, b,
                                                   (short)0, c1, false, false);
    }

    const int ocol = ot * 16 + lm;
    if (rem >= 32) {
        // full 32-row tile: unconditional atomics
#pragma unroll
        for (int v = 0; v < 8; ++v) {
            int ml = v + 8 * half;
            int b0 = row_b[off + mt * 32 + ml];
            int b1 = row_b[off + mt * 32 + 16 + ml];
            atomicAdd(y + (size_t)b0 * OUT_F + ocol, c0[v]);
            atomicAdd(y + (size_t)b1 * OUT_F + ocol, c1[v]);
        }
    } else {
#pragma unroll
        for (int v = 0; v < 8; ++v) {
            int ml = v + 8 * half;
            if (ml < rem) {
                int b0 = row_b[off + mt * 32 + ml];
                atomicAdd(y + (size_t)b0 * OUT_F + ocol, c0[v]);
            }
            if (ml + 16 < rem) {
                int b1 = row_b[off + mt * 32 + 16 + ml];
                atomicAdd(y + (size_t)b1 * OUT_F + ocol, c1[v]);
            }
        }
    }
}

// ---------------------------------------------------------------------------
// Host-side launcher
// ---------------------------------------------------------------------------
extern "C" void kernel_launch(void* const* d_in, const int* in_sizes, int n_in,
                              void* d_out, int out_size, void* d_ws, size_t ws_size,
                              hipStream_t stream) {
    const float* x  = (const float*)d_in[0];
    const float* q  = (const float*)d_in[1];
    const float* Wk = (const float*)d_in[2];
    const float* bk = (const float*)d_in[3];
    const float* V0 = (const float*)d_in[4];
    const float* V1 = (const float*)d_in[5];
    float* y = (float*)d_out;

    char* ws = (char*)d_ws;
    int*   counts  = (int*)(ws + 0);            //  64 ints
    int*   cursor  = (int*)(ws + 256);          //  64 ints (contig w/ counts)
    int*   offsets = (int*)(ws + 512);          //  64 ints
    int*   act_s   = (int*)(ws + 1024);         // 2048 ints
    float* act_w   = (float*)(ws + 1024 + 8192);
    int*   row_b   = (int*)(ws + 1024 + 16384);
    float* row_w   = (float*)(ws + 1024 + 24576);
    float* H       = (float*)(ws + 32768);      // 2048 x 128 f32 = 1 MB

    const int ny = B_SZ * OUT_F;                // 262144
    k_zero<<<(ny + 255) / 256, 256, 0, stream>>>(y, ny, counts, 128);

    k_attn_topk<<<B_SZ, N_SUB, 0, stream>>>(q, Wk, bk, act_s, act_w, counts);

    k_scan_scatter<<<1, 256, 0, stream>>>(counts, offsets, cursor,
                                          act_s, act_w, row_b, row_w);

    k_stageA<<<dim3(SUB_F / 16, B_SZ / 16, N_SUB), 32, 0, stream>>>(
        x, V0, counts, offsets, row_b, row_w, H);

    k_stageB<<<dim3(OUT_F / 16, B_SZ / 32, N_SUB), 32, 0, stream>>>(
        H, V1, counts, offsets, row_b, y);
}